// LightGCNConv_3358664426026
// MI455X (gfx1250) — compile-verified
//
#include <hip/hip_runtime.h>
#include <hip/hip_bf16.h>

// LightGCN 3-layer propagation on MI455X (gfx1250, wave32).
//
// The 25.6MB feature matrix is L2-resident (192MB L2); the hazard is scatter
// atomics (307M f32 adds naively). We bin edges by 16-row destination tiles
// (CSR build with cheap u32 atomics), then one wave owns a 16x64 output tile
// in registers and consumes edges 4-per-K-step with V_WMMA_F32_16X16X4_F32:
//   A = 16x4 one-hot-weighted matrix (edge weight at [dst%16, k])
//   B = 4x16 gathered source features
//   D(16x16) += A x B   -- the WMMA performs the per-edge row-scatter.
// Edge metadata is wave-uniform -> fetched on the scalar (SMEM) pipe via
// readfirstlane-forced uniformity; zero atomics and zero EXEC divergence in
// the hot loop; one store per output element.

#define NN 100000           // N_NODES
#define NE 1600000          // N_EDGES
#define DF 64               // D_FEAT
#define NT (NN / 16)        // 6250 destination row-tiles (exact)
#define PAD 8               // tail slack for unconditional uniform edge loads

typedef __attribute__((ext_vector_type(2))) float v2f;
typedef __attribute__((ext_vector_type(8))) float v8f;

// ---------------------------------------------------------------- utilities
__global__ void zero_u32_kernel(unsigned* __restrict__ p, int n) {
    int i = blockIdx.x * blockDim.x + threadIdx.x;
    if (i < n) p[i] = 0u;
}

__global__ void copy_u32_kernel(unsigned* __restrict__ d, const unsigned* __restrict__ s, int n) {
    int i = blockIdx.x * blockDim.x + threadIdx.x;
    if (i < n) d[i] = s[i];
}

// out = 0.25*(x + h1 + h2 + h3), vectorized 16B.
__global__ void mean4_kernel(float4* __restrict__ out, const float4* __restrict__ x,
                             const float4* __restrict__ h1, const float4* __restrict__ h2,
                             const float4* __restrict__ h3, int n4) {
    int i = blockIdx.x * blockDim.x + threadIdx.x;
    if (i >= n4) return;
    float4 a = x[i], b = h1[i], c = h2[i], d = h3[i];
    float4 r;
    r.x = 0.25f * (a.x + b.x + c.x + d.x);
    r.y = 0.25f * (a.y + b.y + c.y + d.y);
    r.z = 0.25f * (a.z + b.z + c.z + d.z);
    r.w = 0.25f * (a.w + b.w + c.w + d.w);
    out[i] = r;
}

// ------------------------------------------------------------- CSR binning
__global__ void count_kernel(const int* __restrict__ dst, unsigned* __restrict__ cnt) {
    int e = blockIdx.x * blockDim.x + threadIdx.x;
    if (e < NE) atomicAdd(&cnt[((unsigned)dst[e]) >> 4], 1u);
}

// Single-workgroup exclusive scan over NT=6250 bin counts.
__global__ __launch_bounds__(1024) void scan_kernel(const unsigned* __restrict__ cnt,
                                                    unsigned* __restrict__ off) {
    __shared__ unsigned lds[1024];
    const int ITEMS = 7;                       // 1024*7 = 7168 >= 6250
    int t = threadIdx.x;
    unsigned loc[ITEMS];
    unsigned run = 0u;
#pragma unroll
    for (int i = 0; i < ITEMS; ++i) {
        int idx = t * ITEMS + i;
        unsigned v = (idx < NT) ? cnt[idx] : 0u;
        loc[i] = run;                          // thread-local exclusive
        run += v;
    }
    lds[t] = run;
    __syncthreads();
    for (int d = 1; d < 1024; d <<= 1) {       // Hillis-Steele inclusive scan
        unsigned addv = (t >= d) ? lds[t - d] : 0u;
        __syncthreads();
        lds[t] += addv;
        __syncthreads();
    }
    unsigned base = (t > 0) ? lds[t - 1] : 0u;
#pragma unroll
    for (int i = 0; i < ITEMS; ++i) {
        int idx = t * ITEMS + i;
        if (idx < NT) off[idx] = base + loc[i];
    }
    if (t == 0) off[NT] = lds[1023];
}

__global__ void scatter_kernel(const int* __restrict__ src, const int* __restrict__ dst,
                               const float* __restrict__ ew, unsigned* __restrict__ cursor,
                               unsigned* __restrict__ sw_perm, float* __restrict__ w_perm) {
    int e = blockIdx.x * blockDim.x + threadIdx.x;
    if (e >= NE) return;
    int d = dst[e];
    unsigned pos = atomicAdd(&cursor[((unsigned)d) >> 4], 1u);
    // src < 2^17, pack src into [31:4], dst%16 into [3:0]
    sw_perm[pos] = (((unsigned)src[e]) << 4) | ((unsigned)d & 15u);
    w_perm[pos]  = ew[e];
}

// ----------------------------------------------------- WMMA SpMM (hot loop)
// One wave per 16-row dst tile; 16x64 accumulated in four 16x16 f32 WMMA
// C/D tiles; 8 edges (two K-steps, 8 WMMAs) per loop iteration.
__global__ __launch_bounds__(256) void spmm_wmma_kernel(
    const float* __restrict__ Xin, float* __restrict__ Hout,
    const unsigned* __restrict__ off, const unsigned* __restrict__ sw_perm,
    const float* __restrict__ w_perm) {
    const int lane = threadIdx.x & 31;
    const int wave = threadIdx.x >> 5;
    const int bin  = blockIdx.x * 8 + wave;          // wave-uniform
    if (bin >= NT) return;                           // whole-wave exit; EXEC stays full
    const int half = lane >> 4;                      // 0: lanes 0-15, 1: lanes 16-31
    const int m    = lane & 15;

    // Force SGPR-hood so the edge stream rides the scalar memory pipe.
    unsigned p   = (unsigned)__builtin_amdgcn_readfirstlane((int)off[bin]);
    unsigned end = (unsigned)__builtin_amdgcn_readfirstlane((int)off[bin + 1]);

    v8f c0 = {}, c1 = {}, c2 = {}, c3 = {};

    for (; p < end; p += 8) {
        // Uniform fetch of 8 edges (s_load; PAD slack makes tail loads safe).
        unsigned swv[8];
        float    wv[8];
#pragma unroll
        for (int k = 0; k < 8; ++k) {
            unsigned q  = p + (unsigned)k;
            unsigned sv = sw_perm[q];
            float    wf = w_perm[q];
            bool ok = q < end;                       // uniform select (s_cselect)
            swv[k] = ok ? sv : 0u;                   // src=0, dst%16=0
            wv[k]  = ok ? wf : 0.0f;                 // weight 0 -> no contribution
        }

#pragma unroll
        for (int j = 0; j < 2; ++j) {
            const int e0 = j * 4;
            // ISA A-layout K split: lanes 0-15 carry K={0,1}, lanes 16-31 K={2,3}.
            unsigned swx = half ? swv[e0 + 2] : swv[e0 + 0];
            unsigned swy = half ? swv[e0 + 3] : swv[e0 + 1];
            float    wx  = half ? wv[e0 + 2]  : wv[e0 + 0];
            float    wy  = half ? wv[e0 + 3]  : wv[e0 + 1];
            int sx = (int)(swx >> 4), dx = (int)(swx & 15u);
            int sy = (int)(swy >> 4), dy = (int)(swy & 15u);

            // A (16x4 f32, 2 VGPRs): VGPR0 = K0|K2, VGPR1 = K1|K3; M = lane%16.
            v2f a;
            a.x = (dx == m) ? wx : 0.0f;
            a.y = (dy == m) ? wy : 0.0f;

            // B (4x16 f32, 2 VGPRs): mirrored K split, N = lane%16.
            // (f32 B layout not explicit in ISA dump; mirrors documented A split.)
            const float* rx = Xin + (size_t)sx * DF + m;
            const float* ry = Xin + (size_t)sy * DF + m;
            v2f b0, b1, b2, b3;
            b0.x = rx[0];  b0.y = ry[0];
            b1.x = rx[16]; b1.y = ry[16];
            b2.x = rx[32]; b2.y = ry[32];
            b3.x = rx[48]; b3.y = ry[48];

            // D = A x B + C  (neg_a, A, neg_b, B, c_mod, C, reuse_a, reuse_b)
            c0 = __builtin_amdgcn_wmma_f32_16x16x4_f32(false, a, false, b0, (short)0, c0, false, false);
            c1 = __builtin_amdgcn_wmma_f32_16x16x4_f32(false, a, false, b1, (short)0, c1, false, false);
            c2 = __builtin_amdgcn_wmma_f32_16x16x4_f32(false, a, false, b2, (short)0, c2, false, false);
            c3 = __builtin_amdgcn_wmma_f32_16x16x4_f32(false, a, false, b3, (short)0, c3, false, false);
        }
    }

    // C/D layout: VGPR v -> row M=v (lanes 0-15) / M=v+8 (lanes 16-31), N=lane%16.
    const size_t rowbase = (size_t)bin * 16 + (size_t)(half * 8);
#pragma unroll
    for (int v = 0; v < 8; ++v) {
        size_t base = (rowbase + v) * DF + m;
        Hout[base +  0] = c0[v];
        Hout[base + 16] = c1[v];
        Hout[base + 32] = c2[v];
        Hout[base + 48] = c3[v];
    }
}

// ------------------------------------------------------------------- driver
extern "C" void kernel_launch(void* const* d_in, const int* in_sizes, int n_in,
                              void* d_out, int out_size, void* d_ws, size_t ws_size,
                              hipStream_t stream) {
    const float* X   = (const float*)d_in[0];
    const int*   src = (const int*)d_in[1];
    const int*   dst = (const int*)d_in[2];
    const float* ew  = (const float*)d_in[3];
    float* out = (float*)d_out;

    // Workspace layout (all 4-byte aligned): ~90 MB total.
    float*    h1      = (float*)d_ws;
    float*    h2      = h1 + (size_t)NN * DF;
    float*    h3      = h2 + (size_t)NN * DF;
    unsigned* cnt     = (unsigned*)(h3 + (size_t)NN * DF);
    unsigned* off     = cnt + NT;
    unsigned* cursor  = off + NT + 1;
    unsigned* sw_perm = cursor + NT;
    float*    w_perm  = (float*)(sw_perm + NE + PAD);

    const int nf = NN * DF;
    dim3 b256(256);
    dim3 gE((NE + 255) / 256);
    dim3 gT((NT + 255) / 256);
    dim3 gS((NT + 7) / 8);             // 8 waves (one 16-row tile each) per block
    dim3 gM((nf / 4 + 255) / 256);

    // CSR build: count -> scan -> scatter. (Intra-bin edge order comes from
    // atomics; only affects FP association of the sums.)
    zero_u32_kernel<<<gT, b256, 0, stream>>>(cnt, NT);
    count_kernel<<<gE, b256, 0, stream>>>(dst, cnt);
    scan_kernel<<<dim3(1), dim3(1024), 0, stream>>>(cnt, off);
    copy_u32_kernel<<<gT, b256, 0, stream>>>(cursor, off, NT);
    scatter_kernel<<<gE, b256, 0, stream>>>(src, dst, ew, cursor, sw_perm, w_perm);

    // 3 propagation layers (pure stores in epilogue; no RMW).
    spmm_wmma_kernel<<<gS, b256, 0, stream>>>(X,  h1, off, sw_perm, w_perm);
    spmm_wmma_kernel<<<gS, b256, 0, stream>>>(h1, h2, off, sw_perm, w_perm);
    spmm_wmma_kernel<<<gS, b256, 0, stream>>>(h2, h3, off, sw_perm, w_perm);

    // Mean over the 4 embeddings.
    mean4_kernel<<<gM, b256, 0, stream>>>((float4*)out, (const float4*)X,
                                          (const float4*)h1, (const float4*)h2,
                                          (const float4*)h3, nf / 4);
}